// SphericalHarmonicEdgeAttrs_25331717112137
// MI455X (gfx1250) — compile-verified
//
#include <hip/hip_runtime.h>
#include <cstdint>

typedef float     v4f __attribute__((ext_vector_type(4), aligned(4)));
typedef float     v3f __attribute__((ext_vector_type(3), aligned(4)));  // b96 row load
typedef long long v2l __attribute__((ext_vector_type(2), aligned(8)));

#define SQRT3  1.7320508075688772f
#define SQRT5  2.2360679774997896f
#define SQRT15 3.8729833462074170f   // sqrt(3)*sqrt(5), folded constant

// ---------------------------------------------------------------------------
// Main kernel: one thread = 4 consecutive edges, NO tail logic. All loops are
// constant-trip + fully unrolled => all state lives in VGPRs (no scratch).
// Gathers fetch each 12B pos row as ONE global_load_b96 (1 L2 request/row).
// ---------------------------------------------------------------------------
__global__ __launch_bounds__(256) void sh_edge_attrs_main(
    const float* __restrict__ pos,       // [N,3]  small, temporal (L2/WGP$ resident)
    const long long* __restrict__ eidx,  // [2,E]  int64, streamed once (NT)
    const float* __restrict__ shift,     // [E,3]  streamed once (NT)
    float* __restrict__ out,             // edge_vec[E*3] | edge_len[E] | edge_sh[E*9]
    int NQ,                              // number of full quads = E/4
    int E)                               // total edge count (section offsets)
{
    const int q = blockIdx.x * blockDim.x + threadIdx.x;
    if (q >= NQ) return;
    const int e0 = q * 4;                // < 3.2M, 32-bit math throughout

    float* __restrict__ out_vec = out;               // [E,3]
    float* __restrict__ out_len = out + 3 * (size_t)E;
    float* __restrict__ out_sh  = out + 4 * (size_t)E;

    // Keep GL2 fill ~196KB ahead of the streaming consumption.
    {
        const int pa = e0 + 16384;
        if (pa + 4 <= 4 * NQ) {
            __builtin_prefetch(eidx + pa, 0, 0);
            __builtin_prefetch(eidx + (size_t)E + pa, 0, 0);
            __builtin_prefetch(shift + 3 * (size_t)pa, 0, 0);
        }
    }

    // ---- streamed inputs: NT b128 loads --------------------------------
    const v2l* srow = (const v2l*)(eidx + e0);
    const v2l* drow = (const v2l*)(eidx + (size_t)E + e0);
    v2l a = __builtin_nontemporal_load(srow);
    v2l b = __builtin_nontemporal_load(srow + 1);
    v2l c = __builtin_nontemporal_load(drow);
    v2l d = __builtin_nontemporal_load(drow + 1);
    // node ids < 100000 -> truncate to 32-bit immediately (kills 64-bit VALU)
    unsigned si[4] = { (unsigned)a.x, (unsigned)a.y, (unsigned)b.x, (unsigned)b.y };
    unsigned di[4] = { (unsigned)c.x, (unsigned)c.y, (unsigned)d.x, (unsigned)d.y };

    const v4f* sp4 = (const v4f*)(shift + 3 * e0);
    v4f f0 = __builtin_nontemporal_load(sp4 + 0);
    v4f f1 = __builtin_nontemporal_load(sp4 + 1);
    v4f f2 = __builtin_nontemporal_load(sp4 + 2);
    float sx[4] = { f0.x, f0.w, f1.z, f2.y };
    float sy[4] = { f0.y, f1.x, f1.w, f2.z };
    float sz[4] = { f0.z, f1.y, f2.x, f2.w };

    float vx[4], vy[4], vz[4], ln[4], sh[36];

    #pragma unroll
    for (int k = 0; k < 4; ++k) {
        // Gathers: one b96 per endpoint row, temporal RT (pos stays cached).
        const v3f p0 = *(const v3f*)(pos + 3 * si[k]);
        const v3f p1 = *(const v3f*)(pos + 3 * di[k]);
        const float dx = p1.x - p0.x - sx[k];
        const float dy = p1.y - p0.y - sy[k];
        const float dz = p1.z - p0.z - sz[k];
        vx[k] = dx; vy[k] = dy; vz[k] = dz;

        const float d2   = dx * dx + dy * dy + dz * dz;
        const float rinv = __builtin_amdgcn_rsqf(d2);   // v_rsq_f32
        ln[k] = d2 * rinv;                              // |v| = d2 * rsqrt(d2)

        const float x = dx * rinv, y = dy * rinv, z = dz * rinv;
        const float x2 = x * x, y2 = y * y, z2 = z * z;

        sh[9 * k + 0] = 1.0f;
        sh[9 * k + 1] = SQRT3 * x;
        sh[9 * k + 2] = SQRT3 * y;
        sh[9 * k + 3] = SQRT3 * z;
        sh[9 * k + 4] = SQRT15 * x * z;
        sh[9 * k + 5] = SQRT15 * x * y;
        sh[9 * k + 6] = SQRT5 * (y2 - 0.5f * (x2 + z2));
        sh[9 * k + 7] = SQRT15 * y * z;
        sh[9 * k + 8] = 0.5f * SQRT15 * (z2 - x2);
    }

    // ---- streamed outputs: NT b128 stores ------------------------------
    float* vp = out_vec + 3 * e0;        // 48*q bytes -> 16B aligned
    __builtin_nontemporal_store((v4f){vx[0], vy[0], vz[0], vx[1]}, (v4f*)(vp + 0));
    __builtin_nontemporal_store((v4f){vy[1], vz[1], vx[2], vy[2]}, (v4f*)(vp + 4));
    __builtin_nontemporal_store((v4f){vz[2], vx[3], vy[3], vz[3]}, (v4f*)(vp + 8));

    __builtin_nontemporal_store((v4f){ln[0], ln[1], ln[2], ln[3]},
                                (v4f*)(out_len + e0));

    float* op = out_sh + 9 * e0;         // 144*q bytes -> 16B aligned
    #pragma unroll
    for (int j = 0; j < 9; ++j) {
        __builtin_nontemporal_store(
            (v4f){sh[4 * j + 0], sh[4 * j + 1], sh[4 * j + 2], sh[4 * j + 3]},
            (v4f*)(op + 4 * j));
    }
}

// ---------------------------------------------------------------------------
// Tail kernel: at most 3 edges, pure scalar, launched only when E % 4 != 0.
// ---------------------------------------------------------------------------
__global__ void sh_edge_attrs_tail(
    const float* __restrict__ pos,
    const long long* __restrict__ eidx,
    const float* __restrict__ shift,
    float* __restrict__ out,
    int e_start, int E)
{
    const int e = e_start + blockIdx.x * blockDim.x + threadIdx.x;
    if (e >= E) return;

    float* __restrict__ out_vec = out;
    float* __restrict__ out_len = out + 3 * (size_t)E;
    float* __restrict__ out_sh  = out + 4 * (size_t)E;

    const unsigned i0 = (unsigned)eidx[e];
    const unsigned i1 = (unsigned)eidx[(size_t)E + e];
    const v3f p0 = *(const v3f*)(pos + 3 * i0);
    const v3f p1 = *(const v3f*)(pos + 3 * i1);

    const float dx = p1.x - p0.x - shift[3 * e + 0];
    const float dy = p1.y - p0.y - shift[3 * e + 1];
    const float dz = p1.z - p0.z - shift[3 * e + 2];

    const float d2   = dx * dx + dy * dy + dz * dz;
    const float rinv = __builtin_amdgcn_rsqf(d2);
    const float x = dx * rinv, y = dy * rinv, z = dz * rinv;
    const float x2 = x * x, y2 = y * y, z2 = z * z;

    out_vec[3 * e + 0] = dx;
    out_vec[3 * e + 1] = dy;
    out_vec[3 * e + 2] = dz;
    out_len[e] = d2 * rinv;

    float* s = out_sh + 9 * (size_t)e;
    s[0] = 1.0f;
    s[1] = SQRT3 * x;
    s[2] = SQRT3 * y;
    s[3] = SQRT3 * z;
    s[4] = SQRT15 * x * z;
    s[5] = SQRT15 * x * y;
    s[6] = SQRT5 * (y2 - 0.5f * (x2 + z2));
    s[7] = SQRT15 * y * z;
    s[8] = 0.5f * SQRT15 * (z2 - x2);
}

extern "C" void kernel_launch(void* const* d_in, const int* in_sizes, int n_in,
                              void* d_out, int out_size, void* d_ws, size_t ws_size,
                              hipStream_t stream) {
    const float*     pos   = (const float*)d_in[0];      // [N,3] f32
    const long long* eidx  = (const long long*)d_in[1];  // [2,E] int64 (reference dtype)
    const float*     shift = (const float*)d_in[2];      // [E,3] f32
    float*           out   = (float*)d_out;              // vec | len | sh, flat

    const int E   = in_sizes[1] / 2;                     // 3,200,000
    const int NQ  = E / 4;                               // full quads
    const int rem = E - NQ * 4;

    if (NQ > 0) {
        const int blocks = (NQ + 255) / 256;
        sh_edge_attrs_main<<<blocks, 256, 0, stream>>>(pos, eidx, shift, out, NQ, E);
    }
    if (rem > 0) {
        sh_edge_attrs_tail<<<1, 32, 0, stream>>>(pos, eidx, shift, out, NQ * 4, E);
    }
}